// IterativeDetector_84559316123802
// MI455X (gfx1250) — compile-verified
//
#include <hip/hip_runtime.h>
#include <math.h>

typedef float v2f __attribute__((ext_vector_type(2)));
typedef float v8f __attribute__((ext_vector_type(8)));

#define FEAT_D 256

// ---- CDNA5 async global->LDS staging (ASYNCcnt path), with safe fallback ----
#if __has_builtin(__builtin_amdgcn_global_load_async_to_lds_b128) && \
    __has_builtin(__builtin_amdgcn_s_wait_asynccnt)
#define HAVE_ASYNC_LDS 1
typedef int v4i_ __attribute__((vector_size(4 * sizeof(int))));
typedef __attribute__((address_space(1))) v4i_ as1_v4i;
typedef __attribute__((address_space(3))) v4i_ as3_v4i;
__device__ __forceinline__ void copy16(const float* g, float* l) {
  __builtin_amdgcn_global_load_async_to_lds_b128((as1_v4i*)g, (as3_v4i*)l, 0, 0);
}
#define ASYNC_WAIT() __builtin_amdgcn_s_wait_asynccnt(0)
#else
#define HAVE_ASYNC_LDS 0
__device__ __forceinline__ void copy16(const float* g, float* l) {
  *(float4*)l = *(const float4*)g;
}
#define ASYNC_WAIT() ((void)0)
#endif

__device__ __forceinline__ void top3_insert(float d, float& t0, float& t1, float& t2) {
  if (d < t2) {
    if (d < t1) {
      t2 = t1;
      if (d < t0) { t1 = t0; t0 = d; }
      else        { t1 = d; }
    } else {
      t2 = d;
    }
  }
}

// ---------------- Phase A: L2-normalize rows; also emit sum(x_hat^2) per row
__global__ void l2norm_rowsq_kernel(const float* __restrict__ in,
                                    float* __restrict__ out,
                                    float* __restrict__ rowsq,
                                    int nrows) {
  const int lane = threadIdx.x & 31;
  const int w = blockIdx.x * (blockDim.x >> 5) + (threadIdx.x >> 5);
  if (w >= nrows) return;
  const float* r = in + (size_t)w * FEAT_D;
  float v[FEAT_D / 32];
  float s = 0.0f;
  #pragma unroll
  for (int j = 0; j < FEAT_D / 32; ++j) { v[j] = r[lane + 32 * j]; s = fmaf(v[j], v[j], s); }
  #pragma unroll
  for (int off = 16; off > 0; off >>= 1) s += __shfl_xor(s, off, 32);
  const float inv = 1.0f / sqrtf(s);
  float* o = out + (size_t)w * FEAT_D;
  float s2 = 0.0f;
  #pragma unroll
  for (int j = 0; j < FEAT_D / 32; ++j) {
    float x = v[j] * inv;
    o[lane + 32 * j] = x;
    s2 = fmaf(x, x, s2);
  }
  #pragma unroll
  for (int off = 16; off > 0; off >>= 1) s2 += __shfl_xor(s2, off, 32);
  if (lane == 0) rowsq[w] = s2;
}

// ---------------- Phase B: per-query top-3 smallest distances to SUPPORT set
// 8 waves/block, each wave owns a 16-query tile; the block shares a double-
// buffered, async-staged 16-row support tile in LDS. K-loop issues 8-chunk
// load groups followed by 8 chained V_WMMA_F32_16X16X4_F32.
#define WAVES_B 8
#define BSTRIDE 260  // dwords per staged row: 256 + 4 pad -> conflict-free ds_load_b64

// Stage one 16-row support tile (16 x 256 f32) into LDS. blockDim.x == 256.
__device__ __forceinline__ void stage_tile(const float* __restrict__ sfn, int sbase,
                                           float* lbuf, int tid) {
  #pragma unroll
  for (int i = 0; i < 4; ++i) {
    const int f = (i << 8) + tid;   // 0..1023 chunk id (16 rows x 64 chunks of 16B)
    const int row = f >> 6;
    const int ch  = f & 63;
    const float* g = sfn + (size_t)(sbase + row) * FEAT_D + (ch << 2);
    float* l = lbuf + row * BSTRIDE + (ch << 2);
    copy16(g, l);
  }
}

__global__ __launch_bounds__(256) void knn_support_kernel(
    const float* __restrict__ qfn, const float* __restrict__ sfn,
    const float* __restrict__ aq,  const float* __restrict__ bs,
    float* __restrict__ T0, float* __restrict__ T1, float* __restrict__ T2,
    int Q, int S) {
  __shared__ __align__(16) float Bbuf[2][16 * BSTRIDE];
  __shared__ float dstile[WAVES_B][16][17];

  const int tid  = threadIdx.x;
  const int wave = tid >> 5;
  const int lane = tid & 31;
  const int qbase = (blockIdx.x * WAVES_B + wave) * 16;
  const int r = lane & 15;   // A: query row in tile; B: support col in tile
  const int h = lane >> 4;   // K-pair selector (offsets 2h, 2h+1)

  const float* qrow = qfn + (size_t)(qbase + r) * FEAT_D + 2 * h;

  float aqv[8];
  #pragma unroll
  for (int rr = 0; rr < 8; ++rr) aqv[rr] = aq[qbase + rr + 8 * h];

  float t0 = INFINITY, t1 = INFINITY, t2 = INFINITY; // valid in lanes 0..15

  const int NT = S >> 4;
  stage_tile(sfn, 0, Bbuf[0], tid);
  ASYNC_WAIT();
  __syncthreads();

  for (int t = 0; t < NT; ++t) {
    // kick off async staging of the next tile while we compute on this one
    if (t + 1 < NT) stage_tile(sfn, (t + 1) << 4, Bbuf[(t + 1) & 1], tid);

    const float* bfrag = &Bbuf[t & 1][0] + r * BSTRIDE + 2 * h;

    v8f c = {0.f, 0.f, 0.f, 0.f, 0.f, 0.f, 0.f, 0.f};
    #pragma unroll
    for (int g = 0; g < 8; ++g) {
      v2f av[8], bv[8];
      #pragma unroll
      for (int u = 0; u < 8; ++u) {
        const int k = g * 32 + u * 4;
        av[u] = *(const v2f*)(qrow + k);   // global, L0-resident (reused 256x)
        bv[u] = *(const v2f*)(bfrag + k);  // LDS, bank-conflict-free
      }
      #pragma unroll
      for (int u = 0; u < 8; ++u)
        c = __builtin_amdgcn_wmma_f32_16x16x4_f32(false, av[u], false, bv[u],
                                                  (short)0, c, false, false);
    }

    // d = sqrt(max(a_i + b_j - 2*dot, 1e-12)); stage 16x16 tile, then row scans
    const float bsv = bs[(t << 4) + r];
    #pragma unroll
    for (int rr = 0; rr < 8; ++rr) {
      float d2 = fmaxf(aqv[rr] + bsv - 2.0f * c[rr], 1e-12f);
      dstile[wave][rr + 8 * h][r] = sqrtf(d2);
    }
    __syncthreads();
    if (lane < 16) {
      #pragma unroll
      for (int n = 0; n < 16; ++n) top3_insert(dstile[wave][lane][n], t0, t1, t2);
    }
    ASYNC_WAIT();     // next tile fully in LDS (this wave's async ops done)
    __syncthreads();  // ...and everyone else's too; also guards dstile reuse
  }

  if (lane < 16) {
    const int q = qbase + lane;
    T0[q] = t0; T1[q] = t1; T2[q] = t2;
  }
}

// ---------------- Phase C: 19 promotion rounds + final scores (one workgroup)
__global__ __launch_bounds__(1024) void iterate_kernel(
    const float* __restrict__ qfn, const float* __restrict__ aq,
    const float* __restrict__ T0, const float* __restrict__ T1,
    const float* __restrict__ T2,
    float* __restrict__ out, int Q, int out_elems, int n_iters) {
  __shared__ float red[1024];
  __shared__ int   list[256];
  __shared__ int   count;
  const int tid = threadIdx.x;
  const int nj = Q >> 10;                 // rows per thread (8 for Q=8192)

  float t0[8], t1[8], t2[8];
  unsigned unk = (1u << nj) - 1u;
  #pragma unroll
  for (int j = 0; j < 8; ++j) {
    if (j < nj) {
      const int i = tid + 1024 * j;
      t0[j] = T0[i]; t1[j] = T1[i]; t2[j] = T2[i];
    } else { t0[j] = t1[j] = t2[j] = INFINITY; }
  }

  for (int it = 0; it < n_iters - 1; ++it) {
    float sc[8];
    float lm = INFINITY;
    #pragma unroll
    for (int j = 0; j < 8; ++j) {
      sc[j] = (t0[j] + t1[j] + t2[j]) / 3.0f;
      if ((j < nj) && (unk & (1u << j))) lm = fminf(lm, sc[j]);
    }
    red[tid] = lm;
    __syncthreads();
    for (int off = 512; off > 0; off >>= 1) {
      if (tid < off) red[tid] = fminf(red[tid], red[tid + off]);
      __syncthreads();
    }
    const float kth = red[0];
    if (tid == 0) count = 0;
    __syncthreads();

    #pragma unroll
    for (int j = 0; j < 8; ++j) {
      if ((j < nj) && (unk & (1u << j)) && (sc[j] <= kth)) {
        int p = atomicAdd(&count, 1);
        if (p < 256) list[p] = tid + 1024 * j;
        unk &= ~(1u << j);
      }
    }
    __syncthreads();

    const int cnt = min(count, 256);
    for (int cidx = 0; cidx < cnt; ++cidx) {
      const int jc = list[cidx];
      const float* qj = qfn + (size_t)jc * FEAT_D;
      const float ajc = aq[jc];
      #pragma unroll
      for (int j = 0; j < 8; ++j) {
        if (j < nj) {
          const int i = tid + 1024 * j;
          const float* qi = qfn + (size_t)i * FEAT_D;
          float dot = 0.0f;
          for (int k = 0; k < FEAT_D; ++k) dot = fmaf(qi[k], qj[k], dot);
          float d = sqrtf(fmaxf(aq[i] + ajc - 2.0f * dot, 1e-12f));
          top3_insert(d, t0[j], t1[j], t2[j]);
        }
      }
    }
    __syncthreads();
  }

  #pragma unroll
  for (int j = 0; j < 8; ++j) {
    if (j < nj) {
      const int i = tid + 1024 * j;
      out[i] = 1.0f - (t0[j] + t1[j] + t2[j]) / 3.0f;
      if (Q + i < out_elems) out[Q + i] = 1.0f;
    }
  }
}

extern "C" void kernel_launch(void* const* d_in, const int* in_sizes, int n_in,
                              void* d_out, int out_size, void* d_ws, size_t ws_size,
                              hipStream_t stream) {
  (void)n_in; (void)ws_size;
  const float* support = (const float*)d_in[0];
  const float* query   = (const float*)d_in[2];
  const int S = in_sizes[0] / FEAT_D;
  const int Q = in_sizes[2] / FEAT_D;

  float* w = (float*)d_ws;
  size_t off = 0;
  float* sfn = w + off; off += (size_t)S * FEAT_D;   // normalized support
  float* qfn = w + off; off += (size_t)Q * FEAT_D;   // normalized query
  float* bs  = w + off; off += (size_t)S;            // support row sumsq
  float* aq  = w + off; off += (size_t)Q;            // query row sumsq
  float* T0  = w + off; off += (size_t)Q;            // running 3 smallest dists
  float* T1  = w + off; off += (size_t)Q;
  float* T2  = w + off; off += (size_t)Q;

  l2norm_rowsq_kernel<<<(S + 7) / 8, 256, 0, stream>>>(support, sfn, bs, S);
  l2norm_rowsq_kernel<<<(Q + 7) / 8, 256, 0, stream>>>(query, qfn, aq, Q);
  knn_support_kernel<<<(Q + 16 * WAVES_B - 1) / (16 * WAVES_B), 256, 0, stream>>>(
      qfn, sfn, aq, bs, T0, T1, T2, Q, S);
  iterate_kernel<<<1, 1024, 0, stream>>>(qfn, aq, T0, T1, T2,
                                         (float*)d_out, Q, out_size, 20);
}